// ZeroConvPatchAttn_20486994002002
// MI455X (gfx1250) — compile-verified
//
#include <hip/hip_runtime.h>
#include <hip/hip_bf16.h>

typedef float v2f __attribute__((ext_vector_type(2)));
typedef float v8f __attribute__((ext_vector_type(8)));

#define CDIM 768
#define NPF 196   // fine cells per image
#define NPC 49    // coarse cells per image
#define NB  64

// GEMM tiling
#define KC 32                 // K-chunk
#define AS_STRIDE 36          // A tile row stride (64 rows): 36*i mod 64 tiles all banks
#define BS_PSTRIDE 144        // B pair-tile stride per k-pair: 2*144 mod 64 = 32

// ---------------------------------------------------------------------------
// K0: entropy masks -> seq_lengths / cls_loc (to d_out tail) + routing tables
// ---------------------------------------------------------------------------
__global__ void k_mask(const float* __restrict__ ent, float* __restrict__ out_tail,
                       int* __restrict__ fine_dst, int* __restrict__ coarse_dst,
                       int* __restrict__ cls_loc, int* __restrict__ cmask) {
  __shared__ int s_nc[NB];
  __shared__ int s_loc[NB];
  int t = threadIdx.x;
  if (t < NB) {
    const float* e = ent + t * (14 * 14);
    int nc = 0;
    for (int j = 0; j < NPC; ++j) {
      int jy = j / 7, jx = j % 7;
      float m = 0.25f * (e[(2 * jy) * 14 + 2 * jx] + e[(2 * jy) * 14 + 2 * jx + 1] +
                         e[(2 * jy + 1) * 14 + 2 * jx] + e[(2 * jy + 1) * 14 + 2 * jx + 1]);
      int msk = (m < 0.5f) ? 1 : 0;
      cmask[t * NPC + j] = msk;
      nc += msk;
    }
    s_nc[t] = nc;
  }
  __syncthreads();
  if (t == 0) {
    int loc = 0;
    for (int b = 0; b < NB; ++b) {
      int nc = s_nc[b];
      int seq = 1 + 4 * (NPC - nc) + nc;  // 197 - 3*nc
      s_loc[b] = loc;
      cls_loc[b] = loc;
      out_tail[b] = (float)seq;
      out_tail[NB + b] = (float)loc;
      loc += seq;
    }
  }
  __syncthreads();
  if (t < NB) {
    int b = t;
    int nf = 4 * (NPC - s_nc[b]);
    int fr = s_loc[b] + 1;
    for (int f = 0; f < NPF; ++f) {
      int jy = (f / 14) >> 1, jx = (f % 14) >> 1;
      int sel = cmask[b * NPC + jy * 7 + jx] ? 0 : 1;
      fine_dst[b * NPF + f] = sel ? fr++ : -1;
    }
    int cr = s_loc[b] + 1 + nf;
    for (int j = 0; j < NPC; ++j)
      coarse_dst[b * NPC + j] = cmask[b * NPC + j] ? cr++ : -1;
  }
}

// ---------------------------------------------------------------------------
// Fused A-tile loaders.
// MODE 0: plain row-major (ld = CDIM)
// MODE 1: fine 16x16 im2col from x (B,3,224,224)
// MODE 2: coarse patches from 2x2-mean downsampled x
// MODE 3: mini-attn token = pe_base gather + mini_pos
// ---------------------------------------------------------------------------
template <int MODE>
__device__ __forceinline__ float load_a(const float* __restrict__ A0,
                                        const float* __restrict__ A1, int m, int k) {
  if (MODE == 0) {
    return A0[(long)m * CDIM + k];
  } else if (MODE == 1) {
    int b = m / NPF, f = m % NPF;
    int gy = f / 14, gx = f % 14;
    int c = k >> 8, py = (k >> 4) & 15, px = k & 15;
    return A0[(((long)(b * 3 + c) * 224) + gy * 16 + py) * 224 + gx * 16 + px];
  } else if (MODE == 2) {
    int b = m / NPC, j = m % NPC;
    int jy = j / 7, jx = j % 7;
    int c = k >> 8, py = (k >> 4) & 15, px = k & 15;
    int r = (jy * 16 + py) * 2, cc = (jx * 16 + px) * 2;
    const float* xp = A0 + (((long)(b * 3 + c) * 224) + r) * 224 + cc;
    return 0.25f * (xp[0] + xp[1] + xp[224] + xp[225]);
  } else {  // MODE 3
    int m2 = m >> 2, q = m & 3;
    int b = m2 / NPC, j = m2 % NPC;
    int jy = j / 7, jx = j % 7;
    int f = (2 * jy + (q >> 1)) * 14 + (2 * jx + (q & 1));
    return A0[((long)(b * NPF + f)) * CDIM + k] + A1[q * CDIM + k];
  }
}

// ---------------------------------------------------------------------------
// Tiled f32 WMMA GEMM: C[M,N] = A[M,768] * B[768,N] + bias
// Block tile 64x128, 8 waves, each wave 16x64 (4 accumulators), K-chunk 32.
// - B tile stored as k-pairs (Bs[p][n][2]): each B fragment is one aligned
//   ds_load_b64 straight into the even VGPR operand pair.
// - Global->LDS traffic software-pipelined through registers: next chunk's
//   global loads are in flight while the current chunk's WMMAs execute.
// - All fragments of a chunk are batched into register arrays so the DS
//   loads clause together and the 32 WMMAs issue consecutively.
// M % 64 == 0 and N % 128 == 0 for every instance (no guards).
// ---------------------------------------------------------------------------
template <int MODE>
__global__ void k_gemm(const float* __restrict__ A0, const float* __restrict__ A1,
                       const float* __restrict__ Bm, const float* __restrict__ bias,
                       float* __restrict__ Cm, int N) {
  __shared__ float As[64 * AS_STRIDE];                 // 64 x 32 (padded)
  __shared__ float Bs[(KC / 2) * BS_PSTRIDE * 2];      // 16 pairs x 128 n x 2
  const int K = CDIM;
  int tid = threadIdx.x;
  int m0 = blockIdx.y * 64;
  int n0 = blockIdx.x * 128;
  int wave = tid >> 5, lane = tid & 31;
  int wrow = (wave & 3) * 16;
  int wcol = (wave >> 2) * 64;
  int lm = lane & 15;
  int kh = (lane >> 4) * 2;  // upper half-lanes: K+2,K+3 of A / rows+2 of B

  v8f acc[4];
#pragma unroll
  for (int t = 0; t < 4; ++t) {
    float bv = bias[n0 + wcol + t * 16 + lm];
#pragma unroll
    for (int r = 0; r < 8; ++r) acc[t][r] = bv;
  }

  const float* arow = &As[(wrow + lm) * AS_STRIDE + kh];
  const float* bbase = &Bs[(kh >> 1) * BS_PSTRIDE * 2 + (wcol + lm) * 2];

  float a_stage[8];
  float4 b_stage[4];

  // stage chunk k0 from global into registers
  auto load_stage = [&](int k0) {
#pragma unroll
    for (int i = 0; i < 8; ++i) {
      int e = tid + i * 256;
      int row = e >> 5, kk = e & 31;
      a_stage[i] = load_a<MODE>(A0, A1, m0 + row, k0 + kk);
    }
#pragma unroll
    for (int i = 0; i < 4; ++i) {
      int idx = tid + i * 256;           // 1024 float4s over 32x128
      int kk = idx >> 5, c4 = idx & 31;
      b_stage[i] = *(const float4*)(Bm + (long)(k0 + kk) * N + n0 + c4 * 4);
    }
  };
  // commit staged registers to LDS tiles
  auto store_stage = [&]() {
#pragma unroll
    for (int i = 0; i < 8; ++i) {
      int e = tid + i * 256;
      int row = e >> 5, kk = e & 31;
      As[row * AS_STRIDE + kk] = a_stage[i];
    }
#pragma unroll
    for (int i = 0; i < 4; ++i) {
      int idx = tid + i * 256;
      int kk = idx >> 5, c4 = idx & 31;
      float* dst = &Bs[(kk >> 1) * BS_PSTRIDE * 2 + (c4 * 4) * 2 + (kk & 1)];
      dst[0] = b_stage[i].x; dst[2] = b_stage[i].y;
      dst[4] = b_stage[i].z; dst[6] = b_stage[i].w;
    }
  };

  load_stage(0);
  store_stage();
  __syncthreads();

  for (int k0 = 0; k0 < K; k0 += KC) {
    bool has_next = (k0 + KC) < K;
    if (has_next) load_stage(k0 + KC);   // HBM/L2 latency hidden behind compute

    // batch every fragment of this chunk, then run all 32 WMMAs back-to-back
    v2f afr[KC / 4];
#pragma unroll
    for (int p = 0; p < KC / 4; ++p) afr[p] = *(const v2f*)(arow + p * 4);
    v2f bfr[KC / 4][4];
#pragma unroll
    for (int p = 0; p < KC / 4; ++p)
#pragma unroll
      for (int t = 0; t < 4; ++t)
        bfr[p][t] = *(const v2f*)(bbase + (p * 2) * BS_PSTRIDE * 2 + t * 32);
#pragma unroll
    for (int p = 0; p < KC / 4; ++p)
#pragma unroll
      for (int t = 0; t < 4; ++t)
        acc[t] = __builtin_amdgcn_wmma_f32_16x16x4_f32(
            false, afr[p], false, bfr[p][t], (short)0, acc[t], false, false);

    __syncthreads();
    if (has_next) {
      store_stage();
      __syncthreads();
    }
  }

  int rowhalf = (lane >> 4) << 3;  // C/D: vgpr r holds rows r (lo lanes) / r+8 (hi lanes)
#pragma unroll
  for (int t = 0; t < 4; ++t)
#pragma unroll
    for (int r = 0; r < 8; ++r)
      Cm[(long)(m0 + wrow + r + rowhalf) * N + n0 + wcol + t * 16 + lm] = acc[t][r];
}

// ---------------------------------------------------------------------------
// Per-coarse-patch 4x4 attention; one wave (32 lanes) per patch.
// ---------------------------------------------------------------------------
__global__ void k_attn(const float* __restrict__ qkv, float* __restrict__ obuf) {
  int wave = threadIdx.x >> 5, lane = threadIdx.x & 31;
  int m2 = blockIdx.x * 8 + wave;
  if (m2 >= NB * NPC) return;
  const float* base = qkv + (long)m2 * 4 * 2304;
  float s[16];
#pragma unroll
  for (int i = 0; i < 16; ++i) s[i] = 0.f;
  for (int c = lane; c < CDIM; c += 32) {
    float q[4], k[4];
#pragma unroll
    for (int j = 0; j < 4; ++j) {
      q[j] = base[j * 2304 + c];
      k[j] = base[j * 2304 + 768 + c];
    }
#pragma unroll
    for (int i = 0; i < 4; ++i)
#pragma unroll
      for (int j = 0; j < 4; ++j) s[i * 4 + j] += q[i] * k[j];
  }
#pragma unroll
  for (int i = 0; i < 16; ++i)
    for (int off = 16; off >= 1; off >>= 1) s[i] += __shfl_xor(s[i], off, 32);

  const float scale = 0.0360843918f;  // 768^-0.5
  float attn[16];
#pragma unroll
  for (int i = 0; i < 4; ++i) {
    float mx = -1e30f;
#pragma unroll
    for (int j = 0; j < 4; ++j) mx = fmaxf(mx, s[i * 4 + j] * scale);
    float sum = 0.f;
#pragma unroll
    for (int j = 0; j < 4; ++j) {
      attn[i * 4 + j] = expf(s[i * 4 + j] * scale - mx);
      sum += attn[i * 4 + j];
    }
    float inv = 1.0f / sum;
#pragma unroll
    for (int j = 0; j < 4; ++j) attn[i * 4 + j] *= inv;
  }
  for (int c = lane; c < CDIM; c += 32) {
    float v[4];
#pragma unroll
    for (int j = 0; j < 4; ++j) v[j] = base[j * 2304 + 1536 + c];
#pragma unroll
    for (int i = 0; i < 4; ++i) {
      float o = attn[i * 4] * v[0] + attn[i * 4 + 1] * v[1] +
                attn[i * 4 + 2] * v[2] + attn[i * 4 + 3] * v[3];
      obuf[((long)m2 * 4 + i) * CDIM + c] = o;
    }
  }
}

// ---------------------------------------------------------------------------
// Element-wise epilogues
// ---------------------------------------------------------------------------
__global__ void k_mean(const float* __restrict__ proj, float* __restrict__ meanb) {
  long i = (long)blockIdx.x * blockDim.x + threadIdx.x;
  if (i >= (long)NB * NPC * CDIM) return;
  int m2 = (int)(i / CDIM), c = (int)(i % CDIM);
  const float* p = proj + (long)m2 * 4 * CDIM + c;
  meanb[i] = 0.25f * (p[0] + p[CDIM] + p[2 * CDIM] + p[3 * CDIM]);
}

__global__ void k_cls(const float* __restrict__ cls_tok, const float* __restrict__ mini_pos,
                      const int* __restrict__ cls_loc, float* __restrict__ out) {
  int b = blockIdx.x;
  long dst = (long)cls_loc[b] * CDIM;
  for (int c = threadIdx.x; c < CDIM; c += blockDim.x)
    out[dst + c] = cls_tok[c] + mini_pos[c];
}

__global__ void k_scatter_fine(const float* __restrict__ pe, const int* __restrict__ fdst,
                               const float* __restrict__ posf, float* __restrict__ out) {
  long i = (long)blockIdx.x * blockDim.x + threadIdx.x;
  if (i >= (long)NB * NPF * CDIM) return;
  int m = (int)(i / CDIM), c = (int)(i % CDIM);
  int d = fdst[m];
  if (d >= 0) out[(long)d * CDIM + c] = pe[i] + posf[(m % NPF) * CDIM + c];
}

__global__ void k_scatter_coarse(const float* __restrict__ craw, const float* __restrict__ z,
                                 const int* __restrict__ cdst, const float* __restrict__ posc,
                                 float* __restrict__ out) {
  long i = (long)blockIdx.x * blockDim.x + threadIdx.x;
  if (i >= (long)NB * NPC * CDIM) return;
  int m = (int)(i / CDIM), c = (int)(i % CDIM);
  int d = cdst[m];
  if (d >= 0) out[(long)d * CDIM + c] = craw[i] + z[i] + posc[(m % NPC) * CDIM + c];
}

// ---------------------------------------------------------------------------
extern "C" void kernel_launch(void* const* d_in, const int* in_sizes, int n_in,
                              void* d_out, int out_size, void* d_ws, size_t ws_size,
                              hipStream_t stream) {
  const float* x        = (const float*)d_in[0];
  const float* entropy  = (const float*)d_in[1];
  const float* W_pe     = (const float*)d_in[2];
  const float* b_pe     = (const float*)d_in[3];
  const float* qkv_w    = (const float*)d_in[4];
  const float* qkv_b    = (const float*)d_in[5];
  const float* proj_w   = (const float*)d_in[6];
  const float* proj_b   = (const float*)d_in[7];
  const float* mini_pos = (const float*)d_in[8];
  const float* zero_w   = (const float*)d_in[9];
  const float* zero_b   = (const float*)d_in[10];
  const float* cls_tok  = (const float*)d_in[11];
  const float* pos_fine = (const float*)d_in[12];
  const float* pos_coarse = (const float*)d_in[13];
  (void)in_sizes; (void)n_in; (void)ws_size;

  const long MF = (long)NB * NPF;          // 12544
  const long MC = (long)NB * NPC;          // 3136
  float* ws = (float*)d_ws;
  float* pe_base    = ws;                          // MF*768
  float* coarse_raw = pe_base + MF * CDIM;         // MC*768
  float* qkv        = coarse_raw + MC * CDIM;      // MF*2304 (12544 token rows)
  float* obuf       = qkv + MF * 2304;             // MF*768
  float* projbuf    = qkv;                         // reuse qkv region after attn
  float* meanb      = qkv + MF * CDIM;
  float* zbuf       = meanb + MC * CDIM;
  int*   fine_dst   = (int*)(obuf + MF * CDIM);
  int*   coarse_dst = fine_dst + MF;
  int*   cls_loc    = coarse_dst + MC;
  int*   cmask      = cls_loc + NB;

  float* out = (float*)d_out;
  float* out_tail = out + (out_size - 2 * NB);  // [seq_lengths(64) | cls_loc(64)]

  // 0: masks, prefix sums, routing tables
  k_mask<<<1, 256, 0, stream>>>(entropy, out_tail, fine_dst, coarse_dst, cls_loc, cmask);

  // 1: pe_base = fine-patch im2col @ W_pe + b_pe   (12544x768x768)
  k_gemm<1><<<dim3(CDIM / 128, MF / 64), 256, 0, stream>>>(x, nullptr, W_pe, b_pe, pe_base, CDIM);

  // 2: coarse_raw = coarse-patch (2x2-mean) @ W_pe + b_pe   (3136x768x768)
  k_gemm<2><<<dim3(CDIM / 128, MC / 64), 256, 0, stream>>>(x, nullptr, W_pe, b_pe, coarse_raw, CDIM);

  // 3: qkv = (pe_base gather + mini_pos) @ qkv_w + qkv_b   (12544x2304x768)
  k_gemm<3><<<dim3(2304 / 128, MF / 64), 256, 0, stream>>>(pe_base, mini_pos, qkv_w, qkv_b, qkv, 2304);

  // 4: per-patch 4x4 softmax attention
  k_attn<<<(int)(MC / 8), 256, 0, stream>>>(qkv, obuf);

  // 5: proj = o @ proj_w + proj_b   (12544x768x768)
  k_gemm<0><<<dim3(CDIM / 128, MF / 64), 256, 0, stream>>>(obuf, nullptr, proj_w, proj_b, projbuf, CDIM);

  // 6: mean over the 4 tokens of each coarse patch
  long nmean = MC * CDIM;
  k_mean<<<(int)((nmean + 255) / 256), 256, 0, stream>>>(projbuf, meanb);

  // 7: z = mean @ zero_w + zero_b   (3136x768x768)
  k_gemm<0><<<dim3(CDIM / 128, MC / 64), 256, 0, stream>>>(meanb, nullptr, zero_w, zero_b, zbuf, CDIM);

  // 8: ragged assembly
  k_cls<<<NB, 256, 0, stream>>>(cls_tok, mini_pos, cls_loc, out);
  long nfine = MF * CDIM;
  k_scatter_fine<<<(int)((nfine + 255) / 256), 256, 0, stream>>>(pe_base, fine_dst, pos_fine, out);
  k_scatter_coarse<<<(int)((nmean + 255) / 256), 256, 0, stream>>>(coarse_raw, zbuf, coarse_dst, pos_coarse, out);
}